// SampleNeighbors_11690900979981
// MI455X (gfx1250) — compile-verified
//
#include <hip/hip_runtime.h>
#include <hip/hip_bf16.h>

typedef __attribute__((ext_vector_type(2))) float v2f;
typedef __attribute__((ext_vector_type(8))) float v8f;

#define N_PTS   8192
#define K_OUT   16
#define LIST    17        // keep top-17, drop nearest
#define WAVES   8         // 256 threads / 32
#define QPW     16        // queries per wave
#define QPB     (WAVES * QPW)   // 128 queries per block
#define FINF    3.4e38f

// dist(m,n) = ||x_m||^2 + ||q_n||^2 - 2 x_m . q_n
// WMMA f32 16x16x4 with homogeneous trick:
//   A row m  = (x, y, z, ||x||^2)         (db, staged in LDS)
//   B col n  = (-2qx, -2qy, -2qz, 1.0)    (query, in regs)
//   D[m][n]  = ||x_m||^2 - 2 x_m . q_n ;  add ||q_n||^2 per lane.

__global__ void __launch_bounds__(256)
knn17_wmma_kernel(const float* __restrict__ xyz2,   // [B, N, 3] database
                  const float* __restrict__ xyz1,   // [B, N, 3] queries
                  int* __restrict__ out)            // [B, N, 16]
{
    extern __shared__ char smem[];
    v2f*   a0 = (v2f*)smem;                 // (x, y)        N_PTS entries, 64 KB
    v2f*   a1 = a0 + N_PTS;                 // (z, ||x||^2)  N_PTS entries, 64 KB
    float* md = (float*)(smem + 2 * N_PTS * sizeof(v2f));          // [256][17] dists
    int*   mi = (int*)((char*)md + 256 * LIST * sizeof(float));    // [256][17] idxs

    const int b    = blockIdx.y;
    const int tid  = threadIdx.x;
    const int wid  = tid >> 5;
    const int lane = tid & 31;
    const int half = lane >> 4;         // 0: lanes 0-15, 1: lanes 16-31
    const int col  = lane & 15;         // query column within wave tile

    const float* db = xyz2 + (size_t)b * N_PTS * 3;
    const float* qs = xyz1 + (size_t)b * N_PTS * 3;

    // ---- stage database into LDS (SoA float2 pairs matching WMMA A layout) ----
#pragma unroll 4
    for (int i = tid; i < N_PTS; i += 256) {
        float x = db[3 * i + 0];
        float y = db[3 * i + 1];
        float z = db[3 * i + 2];
        v2f p0; p0.x = x; p0.y = y;
        v2f p1; p1.x = z; p1.y = x * x + y * y + z * z;
        a0[i] = p0;
        a1[i] = p1;
    }

    // ---- per-lane sorted top-17 list lives in LDS (dynamic-index friendly) ----
    float* ld = md + (size_t)tid * LIST;
    int*   li = mi + (size_t)tid * LIST;
#pragma unroll
    for (int j = 0; j < LIST; ++j) { ld[j] = FINF; li[j] = 0x7fffffff; }

    __syncthreads();

    // ---- per-lane query setup ----
    const int q  = blockIdx.x * QPB + wid * QPW + col;
    const float qx = qs[3 * q + 0];
    const float qy = qs[3 * q + 1];
    const float qz = qs[3 * q + 2];
    const float yy = qx * qx + qy * qy + qz * qz;

    v2f bop;
    if (half == 0) { bop.x = -2.0f * qx; bop.y = -2.0f * qy; }
    else           { bop.x = -2.0f * qz; bop.y = 1.0f;       }

    float worst    = FINF;        // cached ld[16]
    int   worstIdx = 0x7fffffff;  // cached li[16]

    // ---- sweep database in 16-row chunks: 1 ds_load_b64 + 1 WMMA per chunk.
    //      A-operand fetch is software-pipelined one chunk ahead so the LDS
    //      latency is covered by the WMMA + select work of the previous chunk.
    const v2f* abase = half ? a1 : a0;
    v2f aop = abase[col];                     // chunk 0 operand

#pragma unroll 1
    for (int cb = 0; cb < N_PTS; cb += 16) {
        const int nb = (cb + 16) & (N_PTS - 1);   // wrapped prefetch index
        v2f anext = abase[nb + col];              // issue early; consumed next iter

        v8f acc = {};
        acc = __builtin_amdgcn_wmma_f32_16x16x4_f32(
            /*neg_a=*/false, aop, /*neg_b=*/false, bop,
            /*c_mod=*/(short)0, acc, /*reuse_a=*/false, /*reuse_b=*/false);

        float dd[8];
#pragma unroll
        for (int r = 0; r < 8; ++r) dd[r] = fmaxf(acc[r] + yy, 0.0f);

        // single cheap test on the hot path: anything possibly makes the list?
        float m01 = fminf(dd[0], dd[1]), m23 = fminf(dd[2], dd[3]);
        float m45 = fminf(dd[4], dd[5]), m67 = fminf(dd[6], dd[7]);
        float mall = fminf(fminf(m01, m23), fminf(m45, m67));

        if (mall <= worst) {                       // cold path, rare
            const int rowBase = cb + 8 * half;     // lanes>=16 hold rows M+8
#pragma unroll
            for (int r = 0; r < 8; ++r) {
                float d = dd[r];
                int idx = rowBase + r;
                if (d < worst || (d == worst && idx < worstIdx)) {
                    // shift-insert into sorted LDS list (ascending, ties by idx)
                    int j = LIST - 1;
#pragma unroll 1
                    while (j > 0) {
                        float pd = ld[j - 1];
                        int   pi = li[j - 1];
                        if (pd < d || (pd == d && pi < idx)) break;
                        ld[j] = pd; li[j] = pi;
                        --j;
                    }
                    ld[j] = d; li[j] = idx;
                    worst = ld[LIST - 1]; worstIdx = li[LIST - 1];
                }
            }
        }

        aop = anext;
    }

    __syncthreads();   // lists of both half-lanes must be visible for the merge

    // ---- merge the two half-lane lists per query, emit ranks 1..16 ----
    if (half == 0) {
        const float* dA = md + (size_t)tid * LIST;          // this lane's list
        const float* dB = md + (size_t)(tid + 16) * LIST;   // partner half-lane
        const int*   iA = mi + (size_t)tid * LIST;
        const int*   iB = mi + (size_t)(tid + 16) * LIST;

        int* op = out + ((size_t)b * N_PTS + q) * K_OUT;
        int i0 = 0, i1 = 0;
#pragma unroll 1
        for (int p = 0; p < LIST; ++p) {
            float d0 = dA[i0], d1 = dB[i1];
            int   j0 = iA[i0], j1 = iB[i1];
            bool take0 = (d0 < d1) || (d0 == d1 && j0 < j1);
            int sel = take0 ? j0 : j1;
            if (take0) ++i0; else ++i1;
            if (p > 0) op[p - 1] = sel;          // drop rank-0 (the nearest point)
        }
    }
}

extern "C" void kernel_launch(void* const* d_in, const int* in_sizes, int n_in,
                              void* d_out, int out_size, void* d_ws, size_t ws_size,
                              hipStream_t stream) {
    (void)in_sizes; (void)n_in; (void)d_ws; (void)ws_size; (void)out_size;
    const float* xyz2 = (const float*)d_in[0];   // database [2,8192,3]
    const float* xyz1 = (const float*)d_in[1];   // queries  [2,8192,3]
    int* out = (int*)d_out;                      // [2,8192,16] int32

    dim3 grid(N_PTS / QPB, 2);                   // 64 x 2 blocks
    size_t smem = 2 * N_PTS * sizeof(v2f)        // 128 KB database tiles
                + 256 * LIST * (sizeof(float) + sizeof(int)); // top-17 lists
    knn17_wmma_kernel<<<grid, 256, smem, stream>>>(xyz2, xyz1, out);
}